// Implicit_Func_2989297238463
// MI455X (gfx1250) — compile-verified
//
#include <hip/hip_runtime.h>
#include <hip/hip_bf16.h>

typedef __attribute__((ext_vector_type(2))) float v2f;
typedef __attribute__((ext_vector_type(8))) float v8f;

#define D 64
#define ALPHA 0.5f
#define LN_EPS 1e-5f
#define STRIPS 4   // row strips (16 rows each) processed per wave with cached B frags

// ---------------------------------------------------------------------------
// Kernel 1: WzUx = nf * ( z@W^T + deg * (x@Uw^T + Ub) )   [N,64]
// One wave32 per (strip-group, 16-col tile). B fragments for W and Uw are
// loaded once into registers and reused across STRIPS row strips.
// ---------------------------------------------------------------------------
__global__ void node_gemm_kernel(const float* __restrict__ z,
                                 const float* __restrict__ x,
                                 const float* __restrict__ nf,
                                 const float* __restrict__ W,
                                 const float* __restrict__ Uw,
                                 const float* __restrict__ Ub,
                                 float* __restrict__ wzux,
                                 int N, int ntasks) {
    int task = blockIdx.x * (blockDim.x >> 5) + (threadIdx.x >> 5);
    if (task >= ntasks) return;                  // wave-uniform
    int sgrp = task >> 2;                        // group of STRIPS row strips
    int jt   = task & 3;                         // 16-col tile
    int j0   = jt * 16;
    int lane = threadIdx.x & 31;
    int m    = lane & 15;                        // A row / B col within tile
    int kh   = lane >> 4;                        // K-half selector

    // Cache all B fragments (16 K-blocks for each of W, Uw) in registers.
    v2f bw[16], bu[16];
    #pragma unroll
    for (int kb = 0; kb < 16; ++kb) {
        int k = kb * 4 + 2 * kh;                 // B[k][n] = W[j0+n][k], contiguous pair
        bw[kb] = *(const v2f*)(W  + (size_t)(j0 + m) * D + k);
        bu[kb] = *(const v2f*)(Uw + (size_t)(j0 + m) * D + k);
    }
    float ub = Ub[j0 + m];

    for (int s = 0; s < STRIPS; ++s) {
        int row0 = (sgrp * STRIPS + s) * 16;
        if (row0 >= N) break;                    // wave-uniform
        int arow = row0 + m; if (arow >= N) arow = N - 1;

        v8f accZ = {}; v8f accX = {};
        #pragma unroll
        for (int kb = 0; kb < 16; ++kb) {
            int k = kb * 4 + 2 * kh;
            v2f az = *(const v2f*)(z + (size_t)arow * D + k);
            v2f ax = *(const v2f*)(x + (size_t)arow * D + k);
            accZ = __builtin_amdgcn_wmma_f32_16x16x4_f32(false, az, false, bw[kb],
                                                         (short)0, accZ, false, false);
            accX = __builtin_amdgcn_wmma_f32_16x16x4_f32(false, ax, false, bu[kb],
                                                         (short)0, accX, false, false);
        }

        int j = j0 + m;
        #pragma unroll
        for (int r = 0; r < 8; ++r) {
            int i = row0 + r + 8 * kh;           // C/D layout: VGPR r <-> row r + 8*lane[4]
            if (i < N) {
                float nfv = nf[i];
                float deg = (nfv > 0.0f) ? (1.0f / nfv) : 0.0f;
                wzux[(size_t)i * D + j] = nfv * (accZ[r] + deg * (accX[r] + ub));
            }
        }
    }
}

// ---------------------------------------------------------------------------
// Kernel 2: zero the aggregation buffer.
// ---------------------------------------------------------------------------
__global__ void zero_kernel(float* __restrict__ p, int n) {
    int i = blockIdx.x * blockDim.x + threadIdx.x;
    if (i < n) p[i] = 0.0f;
}

// ---------------------------------------------------------------------------
// Kernel 3: per edge — gather, relu(diff), LayerNorm(D=64), +/- atomic scatter.
// One wave32 per edge; edge descriptors forced uniform (s_load path) via
// readfirstlane; each lane owns cols {2*lane, 2*lane+1} (coalesced b64).
// ---------------------------------------------------------------------------
__global__ void edge_kernel(const float* __restrict__ wzux,
                            const int* __restrict__ row,
                            const int* __restrict__ col,
                            const float* __restrict__ nf,
                            const float* __restrict__ gamma,
                            const float* __restrict__ beta,
                            float* __restrict__ agg,
                            int E) {
    int wave = (blockIdx.x * blockDim.x + threadIdx.x) >> 5;
    if (wave >= E) return;                       // wave-uniform
    wave = __builtin_amdgcn_readfirstlane(wave); // prove uniformity -> scalar loads
    int lane = threadIdx.x & 31;
    int r = __builtin_amdgcn_readfirstlane(row[wave]);
    int c = __builtin_amdgcn_readfirstlane(col[wave]);
    int j = lane * 2;

    v2f a = *(const v2f*)(wzux + (size_t)r * D + j);
    v2f b = *(const v2f*)(wzux + (size_t)c * D + j);
    float e0 = a.x - b.x; e0 = (e0 > 0.0f) ? e0 : 0.0f;
    float e1 = a.y - b.y; e1 = (e1 > 0.0f) ? e1 : 0.0f;

    float s  = e0 + e1;
    float sq = e0 * e0 + e1 * e1;
    #pragma unroll
    for (int off = 16; off > 0; off >>= 1) {
        s  += __shfl_xor(s,  off, 32);
        sq += __shfl_xor(sq, off, 32);
    }
    float mu  = s * (1.0f / 64.0f);
    float var = sq * (1.0f / 64.0f) - mu * mu;   // population variance (jnp.var)
    float inv = rsqrtf(var + LN_EPS);

    float n0 = (e0 - mu) * inv * gamma[j]     + beta[j];
    float n1 = (e1 - mu) * inv * gamma[j + 1] + beta[j + 1];

    float nfr = nf[r];
    float nfc = nf[c];
    atomicAdd(agg + (size_t)r * D + j,      n0 * nfr);
    atomicAdd(agg + (size_t)r * D + j + 1,  n1 * nfr);
    atomicAdd(agg + (size_t)c * D + j,     -n0 * nfc);
    atomicAdd(agg + (size_t)c * D + j + 1, -n1 * nfc);
}

// ---------------------------------------------------------------------------
// Kernel 4: out = -ALPHA * (agg @ W) + (1-ALPHA) * z
// Same tiling; B fragments (W column tiles) cached across STRIPS row strips.
// ---------------------------------------------------------------------------
__global__ void final_gemm_kernel(const float* __restrict__ agg,
                                  const float* __restrict__ W,
                                  const float* __restrict__ z,
                                  float* __restrict__ out,
                                  int N, int ntasks) {
    int task = blockIdx.x * (blockDim.x >> 5) + (threadIdx.x >> 5);
    if (task >= ntasks) return;
    int sgrp = task >> 2;
    int jt   = task & 3;
    int j0   = jt * 16;
    int lane = threadIdx.x & 31;
    int m    = lane & 15;
    int kh   = lane >> 4;

    v2f bw[16];
    #pragma unroll
    for (int kb = 0; kb < 16; ++kb) {
        int k = kb * 4 + 2 * kh;                 // B[k][n] = W[k][j0+n]
        v2f b;
        b.x = W[(size_t)k * D + j0 + m];
        b.y = W[(size_t)(k + 1) * D + j0 + m];
        bw[kb] = b;
    }

    for (int s = 0; s < STRIPS; ++s) {
        int row0 = (sgrp * STRIPS + s) * 16;
        if (row0 >= N) break;
        int arow = row0 + m; if (arow >= N) arow = N - 1;

        v8f acc = {};
        #pragma unroll
        for (int kb = 0; kb < 16; ++kb) {
            int k = kb * 4 + 2 * kh;
            v2f a = *(const v2f*)(agg + (size_t)arow * D + k);
            acc = __builtin_amdgcn_wmma_f32_16x16x4_f32(false, a, false, bw[kb],
                                                        (short)0, acc, false, false);
        }

        int j = j0 + m;
        #pragma unroll
        for (int r = 0; r < 8; ++r) {
            int i = row0 + r + 8 * kh;
            if (i < N) {
                out[(size_t)i * D + j] = -ALPHA * acc[r]
                                       + (1.0f - ALPHA) * z[(size_t)i * D + j];
            }
        }
    }
}

// ---------------------------------------------------------------------------
extern "C" void kernel_launch(void* const* d_in, const int* in_sizes, int n_in,
                              void* d_out, int out_size, void* d_ws, size_t ws_size,
                              hipStream_t stream) {
    const float* z     = (const float*)d_in[0];
    const float* x     = (const float*)d_in[1];
    const int*   ei    = (const int*)  d_in[2];
    const float* nf    = (const float*)d_in[3];
    const float* W     = (const float*)d_in[4];
    const float* Uw    = (const float*)d_in[5];
    const float* Ub    = (const float*)d_in[6];
    const float* gamma = (const float*)d_in[7];
    const float* beta  = (const float*)d_in[8];

    int N = in_sizes[3];          // norm_factor: [N,1]
    int E = in_sizes[2] / 2;      // edge_index: [2,E]
    const int* row = ei;
    const int* col = ei + E;

    float* wzux = (float*)d_out;  // staged here; dead before final kernel rewrites d_out
    float* agg  = (float*)d_ws;   // N*64 fp32 scatter accumulator

    int sgroups = (N + 16 * STRIPS - 1) / (16 * STRIPS);
    int ntasks  = sgroups * 4;    // x 4 col-tiles
    int wpb     = 4;              // waves per block
    int gblk    = (ntasks + wpb - 1) / wpb;

    node_gemm_kernel<<<gblk, 32 * wpb, 0, stream>>>(z, x, nf, W, Uw, Ub, wzux, N, ntasks);

    int nz = N * D;
    zero_kernel<<<(nz + 255) / 256, 256, 0, stream>>>(agg, nz);

    int epb  = 8;                 // edges (waves) per block
    int eblk = (E + epb - 1) / epb;
    edge_kernel<<<eblk, 32 * epb, 0, stream>>>(wzux, row, col, nf, gamma, beta, agg, E);

    final_gemm_kernel<<<gblk, 32 * wpb, 0, stream>>>(agg, W, z, (float*)d_out, N, ntasks);
}